// SNNetwork_71330816852783
// MI455X (gfx1250) — compile-verified
//
#include <hip/hip_runtime.h>
#include <hip/hip_bf16.h>

// ---------------- problem constants ----------------
#define N_IN   1000
#define N_H    3000
#define N_OUT  10
#define N_TOT  4010          // N_IN + N_H + N_OUT
#define L_TOT  3010          // N_H + N_OUT
#define A_DIM  2
#define B_FF   2
#define B_FB   2
#define TAU    10

#define RED    16040         // N_TOT * A_DIM * B_FF  (GEMV reduction length)
#define ROWS   6020          // L_TOT * A_DIM         (GEMV output rows)
#define TILES  377           // ceil(ROWS / 16)
#define KSPLIT 20            // k-dim splits per row tile (deterministic partials)

typedef __attribute__((ext_vector_type(2))) float v2f;
typedef __attribute__((ext_vector_type(4))) float v4f;
typedef __attribute__((ext_vector_type(8))) float v8f;

// =====================================================================
// Kernel 1: temporal filtering.
//   x_ff[(n*2+a)*2+k] = sum_t hist[n,a,9-t] * ff_filter[k,t]
//   x_fb[...]          = same with fb_filter
// =====================================================================
__global__ void snn_prep(const float* __restrict__ hist,
                         const float* __restrict__ ffF,
                         const float* __restrict__ fbF,
                         float* __restrict__ xff,
                         float* __restrict__ xfb) {
    int idx = blockIdx.x * blockDim.x + threadIdx.x;
    if (idx >= RED) return;
    int k  = idx & 1;        // basis index
    int na = idx >> 1;       // n*A + a
    const float* h = hist + na * TAU;
    float sff = 0.f, sfb = 0.f;
#pragma unroll
    for (int t = 0; t < TAU; ++t) {
        float s = h[TAU - 1 - t];            // flipped history
        sff += s * ffF[k * TAU + t];
        sfb += s * fbF[k * TAU + t];
    }
    xff[idx] = sff;
    xfb[idx] = sfb;
}

// =====================================================================
// Kernel 2: the 386-MB streaming GEMV via V_WMMA_F32_16X16X4_F32.
//
// D = A(16x4) * B(4x16) + C.  The vector is broadcast into every row of A
// (lane l holds row l%16, logical K = 2*(l/16)+vgpr); 16 weight rows go
// (transposed) into B (lane l holds column n = l%16, same logical-K map).
// Every row of D then equals the 16 partial dot products.
//
// Weight stream (read-once, 386 MB > 192 MB L2) uses NON-TEMPORAL b128
// loads so it does not thrash L2; the 64 KB vector stays cache-resident.
// Chunk loop unrolled x4 with loads hoisted ahead of the 8 WMMAs so each
// wave keeps 4 weight loads in flight.
// Per-(tile,split) partials land in private slots: bitwise deterministic.
// =====================================================================
__device__ __forceinline__ v8f wmma_step(v8f acc, v4f xv, v4f wv) {
    v2f a0 = { xv.x, xv.y };
    v2f b0 = { wv.x, wv.y };
    acc = __builtin_amdgcn_wmma_f32_16x16x4_f32(false, a0, false, b0,
                                                (short)0, acc, false, false);
    v2f a1 = { xv.z, xv.w };
    v2f b1 = { wv.z, wv.w };
    acc = __builtin_amdgcn_wmma_f32_16x16x4_f32(false, a1, false, b1,
                                                (short)0, acc, false, false);
    return acc;
}

__global__ __launch_bounds__(256)
void snn_gemv_wmma(const float* __restrict__ W,     // (ROWS, RED) row-major
                   const float* __restrict__ x,     // (RED)
                   float* __restrict__ part) {      // (KSPLIT, ROWS)
    const int lane  = threadIdx.x & 31;
    const int wvblk = threadIdx.x >> 5;
    const int gwave = blockIdx.x * 8 + wvblk;
    const int tile  = gwave / KSPLIT;
    const int split = gwave % KSPLIT;
    if (tile >= TILES) return;                 // wave-uniform exit (EXEC stays full)

    const int ncol = lane & 15;                // output row handled by this lane (B column)
    const int half = lane >> 4;                // K half
    const int row  = tile * 16 + ncol;
    const int rowc = row < ROWS ? row : ROWS - 1;   // clamp padding rows (discarded later)
    const float* wrow = W + (size_t)rowc * RED + 4 * half;
    const float* xvec = x + 4 * half;

    // k-chunks of 8 elements (two 16x16x4 WMMAs per chunk)
    const int totalChunks = RED / 8;           // 2005
    const int c0 = (int)(((long long)split       * totalChunks) / KSPLIT);
    const int c1 = (int)(((long long)(split + 1) * totalChunks) / KSPLIT);

    v8f acc = {};
    int c = c0;
    // ---- unrolled x4: 4 NT weight b128 loads in flight, then 8 WMMAs ----
    for (; c + 4 <= c1; c += 4) {
        const float* wp = wrow + c * 8;
        const float* xp = xvec + c * 8;
        v4f w0 = __builtin_nontemporal_load((const v4f*)(wp));
        v4f w1 = __builtin_nontemporal_load((const v4f*)(wp + 8));
        v4f w2 = __builtin_nontemporal_load((const v4f*)(wp + 16));
        v4f w3 = __builtin_nontemporal_load((const v4f*)(wp + 24));
        v4f x0 = *(const v4f*)(xp);
        v4f x1 = *(const v4f*)(xp + 8);
        v4f x2 = *(const v4f*)(xp + 16);
        v4f x3 = *(const v4f*)(xp + 24);
        acc = wmma_step(acc, x0, w0);
        acc = wmma_step(acc, x1, w1);
        acc = wmma_step(acc, x2, w2);
        acc = wmma_step(acc, x3, w3);
    }
    // ---- remainder chunks ----
    for (; c < c1; ++c) {
        v4f wv = __builtin_nontemporal_load((const v4f*)(wrow + c * 8));
        v4f xv = *(const v4f*)(xvec + c * 8);
        acc = wmma_step(acc, xv, wv);
    }

    // D rows are all identical; D[0][n] lives in VGPR0 of lanes 0..15.
    if (half == 0 && row < ROWS)
        part[split * ROWS + row] = acc[0];
}

// =====================================================================
// Kernel 3: finalize — fb_pot, bias, Threefry/Gumbel categorical sampling
// (replicating jax.random.categorical(key(1), ...)), log-softmax, coef dot.
// =====================================================================
__device__ __forceinline__ unsigned rotl32(unsigned x, unsigned d) {
    return (x << d) | (x >> (32u - d));
}

__device__ void threefry2x32(unsigned k0, unsigned k1,
                             unsigned x0, unsigned x1,
                             unsigned& o0, unsigned& o1) {
    const unsigned k2 = k0 ^ k1 ^ 0x1BD11BDAu;
    const unsigned rA[4] = {13u, 15u, 26u, 6u};
    const unsigned rB[4] = {17u, 29u, 16u, 24u};
    x0 += k0; x1 += k1;
#pragma unroll
    for (int i = 0; i < 4; ++i) { x0 += x1; x1 = rotl32(x1, rA[i]); x1 ^= x0; }
    x0 += k1; x1 += k2 + 1u;
#pragma unroll
    for (int i = 0; i < 4; ++i) { x0 += x1; x1 = rotl32(x1, rB[i]); x1 ^= x0; }
    x0 += k2; x1 += k0 + 2u;
#pragma unroll
    for (int i = 0; i < 4; ++i) { x0 += x1; x1 = rotl32(x1, rA[i]); x1 ^= x0; }
    x0 += k0; x1 += k1 + 3u;
#pragma unroll
    for (int i = 0; i < 4; ++i) { x0 += x1; x1 = rotl32(x1, rB[i]); x1 ^= x0; }
    x0 += k1; x1 += k2 + 4u;
#pragma unroll
    for (int i = 0; i < 4; ++i) { x0 += x1; x1 = rotl32(x1, rA[i]); x1 ^= x0; }
    x0 += k2; x1 += k0 + 5u;
    o0 = x0; o1 = x1;
}

// Gumbel sample for flat element i of the (N_H, 3) logits array,
// matching JAX threefry_random_bits counter pairing (n=9000, half=4500).
__device__ float jax_gumbel(int i) {
    const unsigned HALF = (unsigned)(N_H * 3 / 2);       // 4500
    unsigned lo = (unsigned)i;
    bool second = lo >= HALF;
    if (second) lo -= HALF;
    unsigned o0, o1;
    threefry2x32(0u, 1u, lo, lo + HALF, o0, o1);         // key(1) -> (0,1)
    unsigned bits = second ? o1 : o0;
    float u = __uint_as_float((bits >> 9) | 0x3f800000u) - 1.0f;   // [0,1)
    const float tiny = 1.1754943508222875e-38f;          // finfo(f32).tiny
    u = u * (1.0f - tiny) + tiny;
    u = fmaxf(u, tiny);
    return -__logf(-__logf(u));
}

__global__ void snn_finalize(const float* __restrict__ part,   // (KSPLIT, ROWS)
                             const float* __restrict__ xfb,    // (N*A*B_FB)
                             const float* __restrict__ fbw,    // (L, A, B_FB)
                             const float* __restrict__ bias,   // (L, A)
                             const float* __restrict__ input_signal, // (N_IN+N_OUT, A)
                             float* __restrict__ out) {        // (L)
    int j = blockIdx.x * blockDim.x + threadIdx.x;
    if (j >= L_TOT) return;

    float p[A_DIM];
#pragma unroll
    for (int a = 0; a < A_DIM; ++a) {
        // deterministic fixed-order reduction of GEMV partials
        float pf = 0.f;
#pragma unroll
        for (int s = 0; s < KSPLIT; ++s) pf += part[s * ROWS + j * A_DIM + a];
        // fb_pot
        float fb = 0.f;
        int n = N_IN + j;
#pragma unroll
        for (int k = 0; k < B_FB; ++k)
            fb += fbw[(j * A_DIM + a) * B_FB + k] * xfb[(n * A_DIM + a) * B_FB + k];
        p[a] = pf + fb + bias[j * A_DIM + a];   // TEMPERATURE = 1
    }

    float s0, s1;
    if (j < N_H) {
        // categorical over [0, p0, p1] via Gumbel-argmax (first max on ties)
        float best = 0.f + jax_gumbel(j * 3 + 0);
        int samp = 0;
        float v1 = p[0] + jax_gumbel(j * 3 + 1);
        if (v1 > best) { best = v1; samp = 1; }
        float v2 = p[1] + jax_gumbel(j * 3 + 2);
        if (v2 > best) { best = v2; samp = 2; }
        s0 = (samp == 1) ? 1.f : 0.f;
        s1 = (samp == 2) ? 1.f : 0.f;
    } else {
        int r = N_IN + (j - N_H);              // input_signal[N_IN:] rows
        s0 = input_signal[r * A_DIM + 0];
        s1 = input_signal[r * A_DIM + 1];
    }

    // log_softmax over [0, p0, p1]
    float m   = fmaxf(0.f, fmaxf(p[0], p[1]));
    float lse = m + __logf(__expf(0.f - m) + __expf(p[0] - m) + __expf(p[1] - m));
    float c0  = 1.0f - (s0 + s1);
    out[j] = c0 * (0.f - lse) + s0 * (p[0] - lse) + s1 * (p[1] - lse);
}

// =====================================================================
// Host launcher
// =====================================================================
extern "C" void kernel_launch(void* const* d_in, const int* in_sizes, int n_in,
                              void* d_out, int out_size, void* d_ws, size_t ws_size,
                              hipStream_t stream) {
    const float* input_signal = (const float*)d_in[0];   // (1010, 2)
    const float* hist         = (const float*)d_in[1];   // (4010, 2, 10)
    const float* ffw          = (const float*)d_in[2];   // (3010,2,4010,2,2)
    const float* fbw          = (const float*)d_in[3];   // (3010,2,2)
    const float* bias         = (const float*)d_in[4];   // (3010,2)
    const float* ffF          = (const float*)d_in[5];   // (2,10)
    const float* fbF          = (const float*)d_in[6];   // (2,10)

    float* ws   = (float*)d_ws;
    float* xff  = ws;                         // RED floats
    float* xfb  = ws + RED;                   // RED floats
    float* part = ws + 2 * RED;               // KSPLIT*ROWS floats (~470 KB)

    snn_prep<<<(RED + 255) / 256, 256, 0, stream>>>(hist, ffF, fbF, xff, xfb);

    const int totalWaves = TILES * KSPLIT;    // 7540
    const int blocks     = (totalWaves + 7) / 8;
    snn_gemv_wmma<<<blocks, 256, 0, stream>>>(ffw, xff, part);

    snn_finalize<<<(L_TOT + 255) / 256, 256, 0, stream>>>(
        part, xfb, fbw, bias, input_signal, (float*)d_out);
}